// SubAttention_25211458027996
// MI455X (gfx1250) — compile-verified
//
#include <hip/hip_runtime.h>

typedef __attribute__((ext_vector_type(2))) float v2f;
typedef __attribute__((ext_vector_type(8))) float v8f;

#define BATCH 8
#define CIN   128
#define CO    32
#define NHW   65536
#define QKC   64            // 2*CO channels in qk

// ---------------- CDNA5 async global->LDS copy (with safe fallback) ----------------
#if defined(__gfx1250__) && __has_builtin(__builtin_amdgcn_global_load_async_to_lds_b128)
#define HAVE_ASYNC 1
#else
#define HAVE_ASYNC 0
#endif

#if HAVE_ASYNC
typedef int v4i_async __attribute__((vector_size(4 * sizeof(int))));
typedef __attribute__((address_space(1))) v4i_async* g_v4i_ptr;
typedef __attribute__((address_space(3))) v4i_async* l_v4i_ptr;
#endif

__device__ __forceinline__ void copy16_g2l(const float* __restrict__ g, float* l) {
#if HAVE_ASYNC
  __builtin_amdgcn_global_load_async_to_lds_b128((g_v4i_ptr)g, (l_v4i_ptr)l, 0, 0);
#else
  *(float4*)l = *(const float4*)g;
#endif
}

__device__ __forceinline__ void wait_g2l() {
#if HAVE_ASYNC
#if __has_builtin(__builtin_amdgcn_s_wait_asynccnt)
  __builtin_amdgcn_s_wait_asynccnt(0);
#else
  asm volatile("s_wait_asynccnt 0" ::: "memory");
#endif
#endif
}

// ---------------- Kernel 1: per-batch Gram of qk (64x64), split-K ----------------
// partial[b][chunk][64*64] written deterministically (no atomics)
#define K1_NCHUNK 16
#define K1_CHUNK  (NHW / K1_NCHUNK)   // 4096 columns per block
#define K1_KC     64                  // LDS tile depth
#define K1_PAD    68                  // row stride in floats (68 mod 64 = 4 -> conflict-free)

__global__ void __launch_bounds__(256) gram_kernel(const float* __restrict__ qk,
                                                   float* __restrict__ partial) {
  __shared__ float tile[2][QKC * K1_PAD];       // 2 x ~17.4 KB (double buffer)
  const int b    = blockIdx.y;
  const int n0   = blockIdx.x * K1_CHUNK;
  const int tid  = threadIdx.x;
  const int lane = tid & 31;
  const int wave = tid >> 5;                    // 8 waves
  const int r    = lane & 15;
  const int ks   = (lane >> 4) << 1;            // 0 or 2 (K select per half-wave)

  // wave w owns tiles w and w+8 of the 4x4 grid of 16x16 Gram tiles
  const int i0a = (wave >> 2) << 4;             // rows of tile 1: 0 or 16
  const int i0b = i0a + 32;                     // rows of tile 2: 32 or 48
  const int j0  = (wave & 3) << 4;              // cols (shared by both tiles)

  v8f acc0 = {};
  v8f acc1 = {};

  const float* qb = qk + (size_t)b * QKC * NHW;

  auto stage = [&](int buf, int c0) {
    // 64 rows x 16 float4 = 1024 async b128 copies, 4 per thread, coalesced
    for (int idx = tid; idx < QKC * (K1_KC / 4); idx += 256) {
      int ch = idx >> 4;
      int c4 = (idx & 15) << 2;
      copy16_g2l(&qb[(size_t)ch * NHW + n0 + c0 + c4], &tile[buf][ch * K1_PAD + c4]);
    }
  };

  stage(0, 0);
  int cur = 0;
  for (int c0 = 0; c0 < K1_CHUNK; c0 += K1_KC) {
    wait_g2l();
    __syncthreads();                            // tile[cur] fully resident
    if (c0 + K1_KC < K1_CHUNK) stage(cur ^ 1, c0 + K1_KC);  // prefetch next
    const float* tl = &tile[cur][0];
#pragma unroll
    for (int k = 0; k < K1_KC; k += 4) {
      v2f a0, a1, bf;
      a0.x = tl[(i0a + r) * K1_PAD + k + ks];
      a0.y = tl[(i0a + r) * K1_PAD + k + ks + 1];
      a1.x = tl[(i0b + r) * K1_PAD + k + ks];
      a1.y = tl[(i0b + r) * K1_PAD + k + ks + 1];
      bf.x = tl[(j0 + r) * K1_PAD + k + ks];
      bf.y = tl[(j0 + r) * K1_PAD + k + ks + 1];
      acc0 = __builtin_amdgcn_wmma_f32_16x16x4_f32(false, a0, false, bf, (short)0, acc0, false, false);
      acc1 = __builtin_amdgcn_wmma_f32_16x16x4_f32(false, a1, false, bf, (short)0, acc1, false, false);
    }
    __syncthreads();                            // all reads of tile[cur] done
    cur ^= 1;
  }

  float* pb = partial + ((size_t)b * K1_NCHUNK + blockIdx.x) * (QKC * QKC);
  const int mAdd = (lane >> 4) << 3;            // C/D layout: M = v (+8 for upper half-wave)
#pragma unroll
  for (int v = 0; v < 8; ++v) {
    int m = v + mAdd;
    pb[(i0a + m) * QKC + j0 + r] = acc0[v];
    pb[(i0b + m) * QKC + j0 + r] = acc1[v];
  }
}

// ------- Kernel 2: reduce partials -> norms -> attn softmax -> Weff/beff -------
__global__ void __launch_bounds__(256) attn_weff_kernel(const float* __restrict__ partial,
                                                        const float* __restrict__ w_v,
                                                        const float* __restrict__ b_v,
                                                        const float* __restrict__ temperature,
                                                        float* __restrict__ Weff,
                                                        float* __restrict__ beff) {
  __shared__ float gramS[QKC * QKC];   // 16 KB
  __shared__ float norms[QKC];
  __shared__ float attn[CO][CO];
  const int b    = blockIdx.x;
  const int tid  = threadIdx.x;
  const int lane = tid & 31;
  const int wave = tid >> 5;
  const float T  = temperature[0];

  // deterministic split-K reduction
  const float* pb = partial + (size_t)b * K1_NCHUNK * (QKC * QKC);
  for (int idx = tid; idx < QKC * QKC; idx += 256) {
    float s = 0.f;
    for (int ck = 0; ck < K1_NCHUNK; ++ck) s += pb[(size_t)ck * (QKC * QKC) + idx];
    gramS[idx] = s;
  }
  __syncthreads();

  if (tid < QKC) norms[tid] = fmaxf(sqrtf(gramS[tid * QKC + tid]), 1e-12f);
  __syncthreads();

  // q = channels 32..63, k = channels 0..31 -> logits = G[32+c, d]/(|q_c||k_d|) * T
  for (int idx = tid; idx < CO * CO; idx += 256) {
    int c = idx >> 5, d = idx & 31;
    attn[c][d] = gramS[(CO + c) * QKC + d] / (norms[CO + c] * norms[d]) * T;
  }
  __syncthreads();

  // softmax: one wave (32 lanes) per row
  for (int c = wave; c < CO; c += 8) {
    float v = attn[c][lane];
    float m = v;
    for (int off = 16; off > 0; off >>= 1) m = fmaxf(m, __shfl_xor(m, off, 32));
    float e = __expf(v - m);
    float s = e;
    for (int off = 16; off > 0; off >>= 1) s += __shfl_xor(s, off, 32);
    attn[c][lane] = e / s;
  }
  __syncthreads();

  // Weff = attn @ W_v  (32x128), beff = attn @ b_v
  float* Wb = Weff + (size_t)b * CO * CIN;
  for (int idx = tid; idx < CO * CIN; idx += 256) {
    int c = idx >> 7, ci = idx & 127;
    float s = 0.f;
#pragma unroll
    for (int d = 0; d < CO; ++d) s += attn[c][d] * w_v[d * CIN + ci];
    Wb[idx] = s;
  }
  if (tid < CO) {
    float s = 0.f;
#pragma unroll
    for (int d = 0; d < CO; ++d) s += attn[tid][d] * b_v[d];
    beff[b * CO + tid] = s;
  }
}

// -------- Kernel 3: out[b] = Weff[b] @ X[b] + beff[b]  (32x128 x 128xN) --------
#define K3_NT    64          // output columns per tile
#define K3_TILES 8           // tiles per block (amortize Weff load, pipeline X)
#define K3_PAD   68          // X row stride (mod 64 = 4 -> conflict-free)
#define W_PAD    132         // Weff row stride (mod 64 = 4 -> conflict-free A-frags)

__global__ void __launch_bounds__(256) out_gemm_kernel(const float* __restrict__ x,
                                                       const float* __restrict__ Weff,
                                                       const float* __restrict__ beff,
                                                       float* __restrict__ out) {
  __shared__ float Wl[CO * W_PAD];      // ~16.9 KB
  __shared__ float Xl[2][CIN * K3_PAD]; // 2 x ~34.8 KB (double buffer)
  const int b     = blockIdx.y;
  const int nBase = blockIdx.x * (K3_NT * K3_TILES);
  const int tid   = threadIdx.x;
  const int lane  = tid & 31;
  const int wave  = tid >> 5;           // 8 waves -> 2x4 grid of 16x16 output tiles
  const int r     = lane & 15;
  const int ks    = (lane >> 4) << 1;
  const int m0    = (wave >> 2) << 4;   // 0 or 16
  const int nn0   = (wave & 3) << 4;    // 0,16,32,48
  const int mAdd  = (lane >> 4) << 3;

  const float* xb = x + (size_t)b * CIN * NHW;
  const float* Wb = Weff + (size_t)b * CO * CIN;

  // stage Weff (async): 32 rows x 32 float4
  for (int idx = tid; idx < CO * (CIN / 4); idx += 256) {
    int c  = idx >> 5;
    int c4 = (idx & 31) << 2;
    copy16_g2l(&Wb[c * CIN + c4], &Wl[c * W_PAD + c4]);
  }

  auto stageX = [&](int buf, int n0) {
    // 128 rows x 16 float4 = 2048 async b128 copies, 8 per thread, coalesced
    for (int idx = tid; idx < CIN * (K3_NT / 4); idx += 256) {
      int c  = idx >> 4;
      int c4 = (idx & 15) << 2;
      copy16_g2l(&xb[(size_t)c * NHW + n0 + c4], &Xl[buf][c * K3_PAD + c4]);
    }
  };

  stageX(0, nBase);

  float bias[8];
#pragma unroll
  for (int v = 0; v < 8; ++v) bias[v] = beff[b * CO + m0 + v + mAdd];

  float* ob = out + (size_t)b * CO * NHW;
  int cur = 0;
  for (int t = 0; t < K3_TILES; ++t) {
    wait_g2l();
    __syncthreads();                    // Wl (t==0) and Xl[cur] resident
    if (t + 1 < K3_TILES) stageX(cur ^ 1, nBase + (t + 1) * K3_NT);

    const float* xl = &Xl[cur][0];
    v8f acc;
#pragma unroll
    for (int v = 0; v < 8; ++v) acc[v] = bias[v];   // bias pre-loaded into C
#pragma unroll
    for (int k = 0; k < CIN; k += 4) {
      v2f a, bf;
      a.x  = Wl[(m0 + r) * W_PAD + k + ks];
      a.y  = Wl[(m0 + r) * W_PAD + k + ks + 1];
      bf.x = xl[(k + ks) * K3_PAD + nn0 + r];
      bf.y = xl[(k + ks + 1) * K3_PAD + nn0 + r];
      acc = __builtin_amdgcn_wmma_f32_16x16x4_f32(false, a, false, bf, (short)0, acc, false, false);
    }

    const int n0 = nBase + t * K3_NT;
#pragma unroll
    for (int v = 0; v < 8; ++v) {
      int m = m0 + v + mAdd;
      ob[(size_t)m * NHW + n0 + nn0 + r] = acc[v];
    }
    __syncthreads();                    // all reads of Xl[cur] done
    cur ^= 1;
  }
}

// ------------------------------- launcher -------------------------------------
extern "C" void kernel_launch(void* const* d_in, const int* in_sizes, int n_in,
                              void* d_out, int out_size, void* d_ws, size_t ws_size,
                              hipStream_t stream) {
  (void)in_sizes; (void)n_in; (void)out_size; (void)ws_size;
  const float* x    = (const float*)d_in[0];   // [8,128,256,256]
  const float* qk   = (const float*)d_in[1];   // [8,64,256,256]
  const float* w_v  = (const float*)d_in[2];   // [32,128]
  const float* b_v  = (const float*)d_in[3];   // [32]
  const float* temp = (const float*)d_in[4];   // [1]
  float* out = (float*)d_out;                  // [8,32,65536] fp32

  // workspace layout (floats): partial Grams | Weff | beff  (~2.23 MB total)
  float* partial = (float*)d_ws;                                    // 8*16*4096
  float* Weff    = partial + (size_t)BATCH * K1_NCHUNK * QKC * QKC; // 8*32*128
  float* beff    = Weff + (size_t)BATCH * CO * CIN;                 // 8*32

  gram_kernel<<<dim3(K1_NCHUNK, BATCH), 256, 0, stream>>>(qk, partial);
  attn_weff_kernel<<<dim3(BATCH), 256, 0, stream>>>(partial, w_v, b_v, temp, Weff, beff);
  out_gemm_kernel<<<dim3(NHW / (K3_NT * K3_TILES), BATCH), 256, 0, stream>>>(x, Weff, beff, out);
}